// PolymerGNN_TgMono_90718299226216
// MI455X (gfx1250) — compile-verified
//
#include <hip/hip_runtime.h>

#define H 256
#define INF_ 128
#define NEG_SLOPE 0.2f
#define BN_EPS 1e-5f
#define LDSPAD 4

typedef __attribute__((ext_vector_type(2))) float v2f;
typedef __attribute__((ext_vector_type(8))) float v8f;

// -------------------- helpers --------------------

__device__ __forceinline__ bool finitef(float x) {
    return (__float_as_uint(x) & 0x7F800000u) != 0x7F800000u;
}

// float atomic max via ordered-int trick (valid for mixed signs; init with -inf)
__device__ __forceinline__ void atomicMaxFloat(float* addr, float val) {
    if (val >= 0.0f) atomicMax((int*)addr, __float_as_int(val));
    else             atomicMin((unsigned int*)addr, __float_as_uint(val));
}

__device__ __forceinline__ unsigned fkey(float s) {
    unsigned u = __float_as_uint(s);
    return (u & 0x80000000u) ? ~u : (u | 0x80000000u);   // monotonic order-preserving
}

__global__ void fill_kernel(float* __restrict__ p, size_t n, float v) {
    size_t i = (size_t)blockIdx.x * blockDim.x + threadIdx.x;
    size_t st = (size_t)gridDim.x * blockDim.x;
    for (; i < n; i += st) p[i] = v;
}

// -------------------- WMMA f32 GEMM: Y[M,H] = X[M,K]@W[K,H] (+ X2@W2) (+ bias) ---------
// One 16x16 output tile per wave; 8 waves/block cover 8 consecutive n-tiles and share
// the 16-row A tile, staged into LDS via CDNA5 async global->LDS b128 copies.
__global__ __launch_bounds__(256)
void wmma_gemm(const float* __restrict__ X, const float* __restrict__ W,
               const float* __restrict__ X2, const float* __restrict__ W2,
               const float* __restrict__ bias, float* __restrict__ Y,
               int M, int K, int K2) {
    __shared__ float As[16 * (H + LDSPAD)];   // max K = 256

    int wave = threadIdx.x >> 5;
    int lane = threadIdx.x & 31;
    int half = lane >> 4;
    int l = lane & 15;
    int m0 = blockIdx.x * 16;
    int n0 = (blockIdx.y * 8 + wave) * 16;
    if (m0 >= M) return;                      // block-uniform

    unsigned ldsBase = (unsigned)(unsigned long long)(void*)&As[0];

    // async copy of the 16 x Kd A-tile into LDS (16B chunks, ASYNCcnt-tracked)
    auto load_tile = [&](const float* Xsrc, int Kd) {
        int chunksPerRow = Kd >> 2;               // 4 floats per chunk
        int nChunks = chunksPerRow << 4;          // 16 rows
        int ldA = Kd + LDSPAD;
        for (int ch = threadIdx.x; ch < nChunks; ch += 256) {
            int r = ch / chunksPerRow;
            int c4 = (ch - r * chunksPerRow) << 2;
            int row = m0 + r; if (row >= M) row = M - 1;
            const float* g = Xsrc + (size_t)row * Kd + c4;
            unsigned lo = ldsBase + (unsigned)((r * ldA + c4) * 4);
            asm volatile("global_load_async_to_lds_b128 %0, %1, off"
                         :: "v"(lo), "v"(g) : "memory");
        }
        asm volatile("s_wait_asynccnt 0x0" ::: "memory");
        __syncthreads();
    };

    v8f acc = {0.f,0.f,0.f,0.f,0.f,0.f,0.f,0.f};

    {
        load_tile(X, K);
        int ldA = K + LDSPAD;
        const float* Arow = &As[l * ldA];
        for (int k = 0; k < K; k += 4) {
            int kk = k + 2 * half;
            v2f a = { Arow[kk], Arow[kk + 1] };           // ds_load_b64, conflict-free
            const float* wp = W + (size_t)kk * H + n0 + l;
            v2f b = { wp[0], wp[H] };
            acc = __builtin_amdgcn_wmma_f32_16x16x4_f32(false, a, false, b,
                                                        (short)0, acc, false, false);
        }
    }
    if (X2) {
        __syncthreads();                                   // done reading tile 1
        load_tile(X2, K2);
        int ldA = K2 + LDSPAD;
        const float* Arow = &As[l * ldA];
        for (int k = 0; k < K2; k += 4) {
            int kk = k + 2 * half;
            v2f a = { Arow[kk], Arow[kk + 1] };
            const float* wp = W2 + (size_t)kk * H + n0 + l;
            v2f b = { wp[0], wp[H] };
            acc = __builtin_amdgcn_wmma_f32_16x16x4_f32(false, a, false, b,
                                                        (short)0, acc, false, false);
        }
    }

    float bv = bias ? bias[n0 + l] : 0.0f;
    for (int r = 0; r < 8; ++r) {
        int row = m0 + r + 8 * half;
        if (row < M) Y[(size_t)row * H + n0 + l] = acc[r] + bv;
    }
}

// -------------------- per-node dual dot: a_src / a_dst --------------------
__global__ __launch_bounds__(256)
void node_dots(const float* __restrict__ h, const float* __restrict__ att_s,
               const float* __restrict__ att_d, float* __restrict__ a_src,
               float* __restrict__ a_dst, int n) {
    int wave = threadIdx.x >> 5, lane = threadIdx.x & 31;
    int node = blockIdx.x * 8 + wave;
    if (node >= n) return;
    const float* hp = h + (size_t)node * H;
    float s = 0.f, d = 0.f;
    for (int f = lane; f < H; f += 32) { float v = hp[f]; s += v * att_s[f]; d += v * att_d[f]; }
    for (int o = 16; o; o >>= 1) { s += __shfl_xor(s, o, 32); d += __shfl_xor(d, o, 32); }
    if (lane == 0) { a_src[node] = s; a_dst[node] = d; }
}

// -------------------- GAT edge passes (edges + self loops) --------------------
__global__ __launch_bounds__(256)
void gat_edge_max(const int* __restrict__ src, const int* __restrict__ dst,
                  const float* __restrict__ a_src, const float* __restrict__ a_dst,
                  float* __restrict__ ebuf, float* __restrict__ emax, int E, int n) {
    int i = blockIdx.x * blockDim.x + threadIdx.x;
    if (i >= E + n) return;
    int s = (i < E) ? src[i] : (i - E);
    int d = (i < E) ? dst[i] : (i - E);
    float e = a_src[s] + a_dst[d];
    e = (e >= 0.f) ? e : NEG_SLOPE * e;
    ebuf[i] = e;
    atomicMaxFloat(&emax[d], e);
}

__global__ __launch_bounds__(256)
void gat_edge_expsum(const int* __restrict__ dst, float* __restrict__ ebuf,
                     const float* __restrict__ emax, float* __restrict__ denom,
                     int E, int n) {
    int i = blockIdx.x * blockDim.x + threadIdx.x;
    if (i >= E + n) return;
    int d = (i < E) ? dst[i] : (i - E);
    float ee = expf(ebuf[i] - emax[d]);
    ebuf[i] = ee;
    atomicAdd(&denom[d], ee);
}

__global__ __launch_bounds__(256)
void gat_edge_msg(const int* __restrict__ src, const int* __restrict__ dst,
                  const float* __restrict__ ebuf, const float* __restrict__ denom,
                  const float* __restrict__ h, float* __restrict__ out, int E, int n) {
    int wave = threadIdx.x >> 5, lane = threadIdx.x & 31;
    int i = blockIdx.x * 8 + wave;
    if (i >= E + n) return;
    int s = (i < E) ? src[i] : (i - E);
    int d = (i < E) ? dst[i] : (i - E);
    float alpha = ebuf[i] / denom[d];
    const float* hp = h + (size_t)s * H;
    float* op = out + (size_t)d * H;
    for (int f = lane; f < H; f += 32) atomicMaxFloat(&op[f], alpha * hp[f]);
}

// fix non-finite -> 0, add bias, accumulate column stats (blockDim == H)
__global__ __launch_bounds__(256)
void gat_finalize_stats(float* __restrict__ x, const float* __restrict__ gbias,
                        float* __restrict__ colsum, float* __restrict__ colsq,
                        int rows_per_block, int n) {
    int f = threadIdx.x;
    int r0 = blockIdx.x * rows_per_block;
    int rend = min(r0 + rows_per_block, n);
    float bv = gbias[f], s = 0.f, ss = 0.f;
    for (int r = r0; r < rend; ++r) {
        float v = x[(size_t)r * H + f];
        v = finitef(v) ? v : 0.f;
        v += bv;
        x[(size_t)r * H + f] = v;
        s += v; ss += v * v;
    }
    atomicAdd(&colsum[f], s);
    atomicAdd(&colsq[f], ss);
}

__global__ __launch_bounds__(256)
void col_stats(const float* __restrict__ x, float* __restrict__ colsum,
               float* __restrict__ colsq, int rows_per_block, int n) {
    int f = threadIdx.x;
    int r0 = blockIdx.x * rows_per_block;
    int rend = min(r0 + rows_per_block, n);
    float s = 0.f, ss = 0.f;
    for (int r = r0; r < rend; ++r) { float v = x[(size_t)r * H + f]; s += v; ss += v * v; }
    atomicAdd(&colsum[f], s);
    atomicAdd(&colsq[f], ss);
}

__global__ __launch_bounds__(256)
void bn_prelu(float* __restrict__ x, const float* __restrict__ colsum,
              const float* __restrict__ colsq, const float* __restrict__ g,
              const float* __restrict__ b, const float* __restrict__ prelu_a,
              int rows_per_block, int n) {
    int f = threadIdx.x;
    float mean = colsum[f] / (float)n;
    float var  = colsq[f] / (float)n - mean * mean;
    float sc = g[f] * rsqrtf(var + BN_EPS);
    float sh = b[f] - mean * sc;
    float a = prelu_a[0];
    int r0 = blockIdx.x * rows_per_block;
    int rend = min(r0 + rows_per_block, n);
    for (int r = r0; r < rend; ++r) {
        float v = x[(size_t)r * H + f] * sc + sh;
        x[(size_t)r * H + f] = (v >= 0.f) ? v : a * v;
    }
}

// -------------------- SAGE max-aggregation / pool sum --------------------
__global__ __launch_bounds__(256)
void edge_aggmax(const int* __restrict__ src, const int* __restrict__ dst,
                 const float* __restrict__ h, float* __restrict__ agg, int E) {
    int wave = threadIdx.x >> 5, lane = threadIdx.x & 31;
    int i = blockIdx.x * 8 + wave;
    if (i >= E) return;
    const float* hp = h + (size_t)src[i] * H;
    float* op = agg + (size_t)dst[i] * H;
    for (int f = lane; f < H; f += 32) atomicMaxFloat(&op[f], hp[f]);
}

__global__ __launch_bounds__(256)
void edge_nsum(const int* __restrict__ src, const int* __restrict__ dst,
               const float* __restrict__ h, float* __restrict__ nsum, int E) {
    int wave = threadIdx.x >> 5, lane = threadIdx.x & 31;
    int i = blockIdx.x * 8 + wave;
    if (i >= E) return;
    const float* hp = h + (size_t)src[i] * H;
    float* op = nsum + (size_t)dst[i] * H;
    for (int f = lane; f < H; f += 32) atomicAdd(&op[f], hp[f]);
}

__global__ __launch_bounds__(256)
void fix_nonfinite(float* __restrict__ x, size_t n) {
    size_t i = (size_t)blockIdx.x * blockDim.x + threadIdx.x;
    size_t st = (size_t)gridDim.x * blockDim.x;
    for (; i < n; i += st) { float v = x[i]; x[i] = finitef(v) ? v : 0.f; }
}

// -------------------- pooling score --------------------
__global__ __launch_bounds__(256)
void score_kernel(const float* __restrict__ nsum, const float* __restrict__ h2,
                  const float* __restrict__ wrel, const float* __restrict__ brel,
                  const float* __restrict__ wroot, float* __restrict__ score, int n) {
    int wave = threadIdx.x >> 5, lane = threadIdx.x & 31;
    int node = blockIdx.x * 8 + wave;
    if (node >= n) return;
    const float* np = nsum + (size_t)node * H;
    const float* hp = h2 + (size_t)node * H;
    float s = 0.f;
    for (int f = lane; f < H; f += 32) s += np[f] * wrel[f] + hp[f] * wroot[f];
    for (int o = 16; o; o >>= 1) s += __shfl_xor(s, o, 32);
    if (lane == 0) score[node] = s + brel[0];
}

// -------------------- radix-select: exact K-th largest key --------------------
__global__ __launch_bounds__(1024)
void topk_thresh(const float* __restrict__ score, int n, int k, unsigned* __restrict__ outT) {
    __shared__ unsigned hist[256];
    __shared__ unsigned s_prefix, s_remaining;
    if (threadIdx.x == 0) { s_prefix = 0u; s_remaining = (unsigned)k; }
    for (int pass = 0; pass < 4; ++pass) {
        int shift = 24 - 8 * pass;
        for (int i = threadIdx.x; i < 256; i += blockDim.x) hist[i] = 0u;
        __syncthreads();
        unsigned maskKnown = pass ? (0xFFFFFFFFu << (shift + 8)) : 0u;
        unsigned prefix = s_prefix;
        for (int i = threadIdx.x; i < n; i += blockDim.x) {
            unsigned u = fkey(score[i]);
            if ((u & maskKnown) == prefix) atomicAdd(&hist[(u >> shift) & 0xFF], 1u);
        }
        __syncthreads();
        if (threadIdx.x == 0) {
            unsigned rem = s_remaining;
            int bin = 255;
            for (; bin > 0; --bin) {
                unsigned c = hist[bin];
                if (rem <= c) break;
                rem -= c;
            }
            s_remaining = rem;
            s_prefix = prefix | ((unsigned)bin << shift);
        }
        __syncthreads();
    }
    if (threadIdx.x == 0) outT[0] = s_prefix;
}

__global__ __launch_bounds__(256)
void embed_max(const float* __restrict__ h2, const float* __restrict__ score,
               const unsigned* __restrict__ T, float* __restrict__ embed, int n) {
    int wave = threadIdx.x >> 5, lane = threadIdx.x & 31;
    int node = blockIdx.x * 8 + wave;
    if (node >= n) return;
    float sc = score[node];
    if (fkey(sc) < T[0]) return;
    float t = tanhf(sc);
    const float* hp = h2 + (size_t)node * H;
    for (int f = lane; f < H; f += 32) atomicMaxFloat(&embed[f], t * hp[f]);
}

// -------------------- final MLP head (single block, 256 threads) --------------------
__global__ __launch_bounds__(256)
void head_kernel(const float* __restrict__ embedA, const float* __restrict__ embedG,
                 const float* __restrict__ addf,
                 const float* __restrict__ fc1w, const float* __restrict__ fc1b,
                 const float* __restrict__ prelu_a,
                 const float* __restrict__ fc2w, const float* __restrict__ fc2b,
                 const float* __restrict__ multw, const float* __restrict__ multb,
                 float* __restrict__ out) {
    __shared__ float z[2 * H + 8];
    __shared__ float red1[256], red2[256];
    int t = threadIdx.x;
    z[t] = embedA[t];
    z[H + t] = embedG[t];
    if (t < 8) z[2 * H + t] = addf[t];
    __syncthreads();
    float acc = fc1b[t];
    for (int i = 0; i < 2 * H + 8; ++i) acc += z[i] * fc1w[(size_t)i * H + t];
    float a = prelu_a[0];
    acc = (acc >= 0.f) ? acc : a * acc;
    red1[t] = acc * fc2w[t];
    red2[t] = acc * multw[t];
    __syncthreads();
    for (int s = 128; s; s >>= 1) {
        if (t < s) { red1[t] += red1[t + s]; red2[t] += red2[t + s]; }
        __syncthreads();
    }
    if (t == 0) {
        float pred = red1[0] + fc2b[0];
        float factor = tanhf(red2[0] + multb[0]);
        out[0] = expf(pred) * factor;
    }
}

// -------------------- host orchestration --------------------

extern "C" void kernel_launch(void* const* d_in, const int* in_sizes, int n_in,
                              void* d_out, int out_size, void* d_ws, size_t ws_size,
                              hipStream_t stream) {
    // inputs in setup_inputs() dict order (params flattened in insertion order)
    const float* Ax   = (const float*)d_in[0];
    const int*   Aei  = (const int*)  d_in[1];
    const float* Gx   = (const float*)d_in[2];
    const int*   Gei  = (const int*)  d_in[3];
    const float* addf = (const float*)d_in[4];
    const float* gat_w     = (const float*)d_in[5];
    const float* gat_att_s = (const float*)d_in[6];
    const float* gat_att_d = (const float*)d_in[7];
    const float* gat_b     = (const float*)d_in[8];
    const float* bn1_g     = (const float*)d_in[9];
    const float* bn1_b     = (const float*)d_in[10];
    const float* prelu1    = (const float*)d_in[11];
    const float* sage_wl   = (const float*)d_in[12];
    const float* sage_bl   = (const float*)d_in[13];
    const float* sage_wr   = (const float*)d_in[14];
    const float* bn2_g     = (const float*)d_in[15];
    const float* bn2_b     = (const float*)d_in[16];
    const float* prelu2    = (const float*)d_in[17];
    const float* pool_wrel = (const float*)d_in[18];
    const float* pool_brel = (const float*)d_in[19];
    const float* pool_wroot= (const float*)d_in[20];
    const float* fc1_w     = (const float*)d_in[21];
    const float* fc1_b     = (const float*)d_in[22];
    const float* prelu_fc  = (const float*)d_in[23];
    const float* fc2_w     = (const float*)d_in[24];
    const float* fc2_b     = (const float*)d_in[25];
    const float* mult_w    = (const float*)d_in[26];
    const float* mult_b    = (const float*)d_in[27];

    const int Nn = in_sizes[0] / INF_;     // 20000
    const int E  = in_sizes[1] / 2;        // 320000
    const int EN = E + Nn;
    const int KPOOL = (Nn + 1) / 2;
    const size_t NB = (size_t)Nn * H;

    float* w = (float*)d_ws;
    float* bufH   = w;                       // [Nn,H] GAT hidden h
    float* bufH1  = bufH  + NB;              // [Nn,H] GAT out -> SAGE in
    float* bufAgg = bufH1 + NB;              // [Nn,H] agg, later nsum
    float* bufH2  = bufAgg + NB;             // [Nn,H] out2
    float* emax   = bufH2 + NB;              // [Nn]
    float* denom  = emax  + Nn;              // [Nn]
    float* asrc   = denom + Nn;              // [Nn]
    float* adst   = asrc  + Nn;              // [Nn]
    float* score  = adst  + Nn;              // [Nn]
    float* ebuf   = score + Nn;              // [E+Nn]
    float* colsum = ebuf  + EN;              // [H]
    float* colsq  = colsum + H;              // [H]
    float* embedA = colsq  + H;              // [H]
    float* embedG = embedA + H;              // [H]
    unsigned* topT = (unsigned*)(embedG + H);

    const float NEGINF = -__builtin_huge_valf();
    const int ROWS = 64;
    const int statBlocks = (Nn + ROWS - 1) / ROWS;

    auto run_graph = [&](const float* x, const int* ei, float* embed) {
        const int* src = ei;
        const int* dst = ei + E;

        // h = x @ gat_w   (WMMA f32, async-LDS staged A)
        wmma_gemm<<<dim3(Nn / 16, H / 128), 256, 0, stream>>>(
            x, gat_w, nullptr, nullptr, nullptr, bufH, Nn, INF_, 0);
        node_dots<<<(Nn + 7) / 8, 256, 0, stream>>>(bufH, gat_att_s, gat_att_d, asrc, adst, Nn);

        fill_kernel<<<512, 256, 0, stream>>>(emax, Nn, NEGINF);
        fill_kernel<<<512, 256, 0, stream>>>(denom, Nn, 0.f);
        fill_kernel<<<2048, 256, 0, stream>>>(bufH1, NB, NEGINF);

        gat_edge_max<<<(EN + 255) / 256, 256, 0, stream>>>(src, dst, asrc, adst, ebuf, emax, E, Nn);
        gat_edge_expsum<<<(EN + 255) / 256, 256, 0, stream>>>(dst, ebuf, emax, denom, E, Nn);
        gat_edge_msg<<<(EN + 7) / 8, 256, 0, stream>>>(src, dst, ebuf, denom, bufH, bufH1, E, Nn);

        fill_kernel<<<1, 256, 0, stream>>>(colsum, H, 0.f);
        fill_kernel<<<1, 256, 0, stream>>>(colsq, H, 0.f);
        gat_finalize_stats<<<statBlocks, 256, 0, stream>>>(bufH1, gat_b, colsum, colsq, ROWS, Nn);
        bn_prelu<<<statBlocks, 256, 0, stream>>>(bufH1, colsum, colsq, bn1_g, bn1_b, prelu1, ROWS, Nn);

        // SAGE max-aggregation
        fill_kernel<<<2048, 256, 0, stream>>>(bufAgg, NB, NEGINF);
        edge_aggmax<<<(E + 7) / 8, 256, 0, stream>>>(src, dst, bufH1, bufAgg, E);
        fix_nonfinite<<<2048, 256, 0, stream>>>(bufAgg, NB);

        // out2 = agg@wl + h1@wr + bl  (dual WMMA f32)
        wmma_gemm<<<dim3(Nn / 16, H / 128), 256, 0, stream>>>(
            bufAgg, sage_wl, bufH1, sage_wr, sage_bl, bufH2, Nn, H, H);
        fill_kernel<<<1, 256, 0, stream>>>(colsum, H, 0.f);
        fill_kernel<<<1, 256, 0, stream>>>(colsq, H, 0.f);
        col_stats<<<statBlocks, 256, 0, stream>>>(bufH2, colsum, colsq, ROWS, Nn);
        bn_prelu<<<statBlocks, 256, 0, stream>>>(bufH2, colsum, colsq, bn2_g, bn2_b, prelu2, ROWS, Nn);

        // SAGPooling
        float* nsum = bufAgg;   // agg no longer needed
        fill_kernel<<<2048, 256, 0, stream>>>(nsum, NB, 0.f);
        edge_nsum<<<(E + 7) / 8, 256, 0, stream>>>(src, dst, bufH2, nsum, E);
        score_kernel<<<(Nn + 7) / 8, 256, 0, stream>>>(nsum, bufH2, pool_wrel, pool_brel,
                                                       pool_wroot, score, Nn);
        topk_thresh<<<1, 1024, 0, stream>>>(score, Nn, KPOOL, topT);
        fill_kernel<<<1, 256, 0, stream>>>(embed, H, NEGINF);
        embed_max<<<(Nn + 7) / 8, 256, 0, stream>>>(bufH2, score, topT, embed, Nn);
    };

    run_graph(Ax, Aei, embedA);
    run_graph(Gx, Gei, embedG);

    head_kernel<<<1, 256, 0, stream>>>(embedA, embedG, addf, fc1_w, fc1_b, prelu_fc,
                                       fc2_w, fc2_b, mult_w, mult_b, (float*)d_out);
    (void)n_in; (void)out_size; (void)ws_size;
}